// DDSCTransformer_36687610642760
// MI455X (gfx1250) — compile-verified
//
#include <hip/hip_runtime.h>
#include <hip/hip_bf16.h>

// ---------------- model dims ----------------
#define BB   4
#define INC  6
#define CC   128
#define HH   8
#define CH   (CC * HH)   // 1024
#define DD   6
#define FFD  (4 * CC)    // 512
#define LL   1024
#define EPS  1e-5f

typedef __attribute__((ext_vector_type(2))) float v2f;
typedef __attribute__((ext_vector_type(8))) float v8f;

// =====================================================================
// Batched WMMA fp32 GEMM, BMx4 register-blocked:
//   each wave computes a (16*BM)x64 tile of C = scale * A*B (+bias)(+relu)
// A row-major [M,K] (or TRANSA: stored [K,M]); B row-major [K,N].
// V_WMMA_F32_16X16X4_F32 fragment layouts per CDNA5 ISA 7.12.2:
//   A 16x4 : VGPR0 = A[m][k+2*hi], VGPR1 = A[m][k+2*hi+1] (m = lane&15, hi = lane>>4)
//   B 4x16 : VGPR0 = B[k+2*hi][n], VGPR1 = B[k+2*hi+1][n] (n = lane&15)
//   C 16x16: VGPR j = C[j+8*hi][n]
// Per K=4 step: BM A-frags + 4 B-frags feed 4*BM WMMAs
//   (BM=4 -> ~0.75 vmem/WMMA; BM=2 -> ~1.25 vmem/WMMA, more waves).
// =====================================================================
template <int TRANSA, int BM>
__global__ __launch_bounds__(256) void wmma_gemm_f32(
    const float* __restrict__ A, const float* __restrict__ B,
    float* __restrict__ Cptr, const float* __restrict__ bias,
    int M, int N, int K, int lda, int ldb, int ldc,
    long long sA, long long sB, long long sC,
    int batch, float scale, int relu)
{
  const int lane = threadIdx.x & 31;
  const int wave = threadIdx.x >> 5;
  const int tilesN = N >> 6;              // 64-wide wave tiles
  const int tilesM = M / (16 * BM);       // (16*BM)-tall wave tiles
  const int tilesPer = tilesM * tilesN;
  int gw = blockIdx.x * 8 + wave;
  if (gw >= batch * tilesPer) return;     // whole-wave exit: EXEC stays all-1 for WMMA
  int b  = gw / tilesPer;
  int t  = gw - b * tilesPer;
  int tm = t / tilesN;
  int tn = t - tm * tilesN;

  const int hi = lane >> 4;
  const int lo = lane & 15;

  const float* Ab = A + (long long)b * sA;
  const float* Bb = B + (long long)b * sB;

  const float* ap[BM];
#pragma unroll
  for (int i = 0; i < BM; ++i) {
    if (TRANSA)                           // A stored [K,M]
      ap[i] = Ab + (long long)(2 * hi) * lda + (tm * (16 * BM) + 16 * i + lo);
    else                                  // A stored [M,K]
      ap[i] = Ab + (long long)(tm * (16 * BM) + 16 * i + lo) * lda + 2 * hi;
  }
  const float* bp = Bb + (long long)(2 * hi) * ldb + (tn * 64 + lo);

  v8f acc[BM][4];
#pragma unroll
  for (int i = 0; i < BM; ++i)
#pragma unroll
    for (int j = 0; j < 4; ++j)
      acc[i][j] = (v8f){0.f, 0.f, 0.f, 0.f, 0.f, 0.f, 0.f, 0.f};

  for (int k = 0; k < K; k += 4) {
    v2f a[BM], bv[4];
#pragma unroll
    for (int i = 0; i < BM; ++i) {
      if (TRANSA) { a[i].x = ap[i][0]; a[i].y = ap[i][lda]; ap[i] += 4LL * lda; }
      else        { a[i].x = ap[i][0]; a[i].y = ap[i][1];   ap[i] += 4; }
    }
#pragma unroll
    for (int j = 0; j < 4; ++j) { bv[j].x = bp[j * 16]; bv[j].y = bp[j * 16 + ldb]; }
    bp += 4LL * ldb;
#pragma unroll
    for (int j = 0; j < 4; ++j)
#pragma unroll
      for (int i = 0; i < BM; ++i)
        acc[i][j] = __builtin_amdgcn_wmma_f32_16x16x4_f32(false, a[i], false, bv[j],
                                                          (short)0, acc[i][j], false, false);
  }

  float* Cb = Cptr + (long long)b * sC + (long long)(tm * (16 * BM)) * ldc + tn * 64;
#pragma unroll
  for (int i = 0; i < BM; ++i) {
#pragma unroll
    for (int j = 0; j < 4; ++j) {
#pragma unroll
      for (int r = 0; r < 8; ++r) {
        int row = i * 16 + r + 8 * hi;
        float v = acc[i][j][r] * scale;
        if (bias) v += bias[tm * (16 * BM) + row];
        if (relu) v = v > 0.f ? v : 0.f;
        Cb[(long long)row * ldc + j * 16 + lo] = v;
      }
    }
  }
}

// =====================================================================
// Encoder DDSC: pointwise 6->128 + softmax-gated depthwise convs (3,15)
// =====================================================================
__global__ __launch_bounds__(256) void encoder_ddsc(
    const float* __restrict__ x, const float* __restrict__ pw,
    const float* __restrict__ dwa, const float* __restrict__ dwb,
    const float* __restrict__ gate, float* __restrict__ out)
{
  __shared__ float t[LL];
  const int b = blockIdx.x / CC;
  const int o = blockIdx.x % CC;
  const float* xb = x + (long long)b * INC * LL;
  for (int l = threadIdx.x; l < LL; l += blockDim.x) {
    float s = 0.f;
#pragma unroll
    for (int c = 0; c < INC; ++c) s += pw[o * INC + c] * xb[c * LL + l];
    t[l] = s;
  }
  __syncthreads();
  float e0 = __expf(gate[0]), e1 = __expf(gate[1]);
  float g0 = e0 / (e0 + e1), g1 = e1 / (e0 + e1);
  const float* wa = dwa + o * 3;
  const float* wb = dwb + o * 15;
  float* ob = out + ((long long)b * CC + o) * LL;
  for (int l = threadIdx.x; l < LL; l += blockDim.x) {
    float sa = 0.f;
#pragma unroll
    for (int j = 0; j < 3; ++j) { int i = l + j - 1; if (i >= 0 && i < LL) sa += wa[j] * t[i]; }
    float sb = 0.f;
#pragma unroll
    for (int j = 0; j < 15; ++j) { int i = l + j - 7; if (i >= 0 && i < LL) sb += wb[j] * t[i]; }
    ob[l] = g0 * sa + g1 * sb;
  }
}

// =====================================================================
// In-place softmax-gated dual depthwise conv over rows of [B*CH, L].
// Row staged into LDS via CDNA5 async global->LDS (ASYNCcnt) path:
//   global_load_async_to_lds_b128 vdst(lds-offset), vaddr64, off  (GV mode)
// LDS offset = low 32 bits of the flat shared-memory pointer (ISA 10.2:
// LDS aperture address maps addr[31:0] to the LDS offset).
// =====================================================================
__global__ __launch_bounds__(256) void dw_conv_rows(
    float* __restrict__ buf, const float* __restrict__ dwa,
    const float* __restrict__ dwb, const float* __restrict__ gate)
{
  __shared__ float t[LL];
  const int row = blockIdx.x;        // b*CH + ch
  const int ch  = row % CH;
  float* rb = buf + (long long)row * LL;

  // each of the 256 lanes pulls one 16B chunk: 256 * 4 floats = 1024
  {
    unsigned lds_off = (unsigned)(unsigned long long)(&t[threadIdx.x * 4]);
    unsigned long long ga = (unsigned long long)(rb + threadIdx.x * 4);
    asm volatile("global_load_async_to_lds_b128 %0, %1, off"
                 :: "v"(lds_off), "v"(ga) : "memory");
    asm volatile("s_wait_asynccnt 0x0" ::: "memory");
  }
  __syncthreads();

  float e0 = __expf(gate[0]), e1 = __expf(gate[1]);
  float g0 = e0 / (e0 + e1), g1 = e1 / (e0 + e1);
  const float* wa = dwa + ch * 3;
  const float* wb = dwb + ch * 15;
  for (int l = threadIdx.x; l < LL; l += blockDim.x) {
    float sa = 0.f;
#pragma unroll
    for (int j = 0; j < 3; ++j) { int i = l + j - 1; if (i >= 0 && i < LL) sa += wa[j] * t[i]; }
    float sb = 0.f;
#pragma unroll
    for (int j = 0; j < 15; ++j) { int i = l + j - 7; if (i >= 0 && i < LL) sb += wb[j] * t[i]; }
    rb[l] = g0 * sa + g1 * sb;
  }
}

// =====================================================================
// Softmax over the KEY axis (rows) of s[h][kl][ql] for each column ql.
// =====================================================================
__global__ __launch_bounds__(256) void softmax_over_rows(float* __restrict__ s)
{
  __shared__ float red[16][17];
  const int tx = threadIdx.x & 15;
  const int ty = threadIdx.x >> 4;
  const int colTiles = LL >> 4;
  const int h  = blockIdx.x / colTiles;
  const int ct = blockIdx.x % colTiles;
  float* base = s + (long long)h * LL * LL + ct * 16 + tx;

  float m = -3.4e38f;
  for (int kl = ty; kl < LL; kl += 16) m = fmaxf(m, base[(long long)kl * LL]);
  red[ty][tx] = m;
  __syncthreads();
  if (ty == 0) {
    float mm = red[0][tx];
#pragma unroll
    for (int j = 1; j < 16; ++j) mm = fmaxf(mm, red[j][tx]);
    red[0][tx] = mm;
  }
  __syncthreads();
  m = red[0][tx];
  __syncthreads();

  float sum = 0.f;
  for (int kl = ty; kl < LL; kl += 16) {
    long long idx = (long long)kl * LL;
    float e = __expf(base[idx] - m);
    base[idx] = e;
    sum += e;
  }
  red[ty][tx] = sum;
  __syncthreads();
  if (ty == 0) {
    float ss = 0.f;
#pragma unroll
    for (int j = 0; j < 16; ++j) ss += red[j][tx];
    red[0][tx] = 1.f / ss;
  }
  __syncthreads();
  float inv = red[0][tx];
  for (int kl = ty; kl < LL; kl += 16) base[(long long)kl * LL] *= inv;
}

// =====================================================================
// dst = InstanceNorm(y + r) over length, affine gamma/beta per channel.
// =====================================================================
__global__ __launch_bounds__(256) void inorm_res(
    const float* __restrict__ y, const float* __restrict__ r,
    float* __restrict__ dst, const float* __restrict__ gamma,
    const float* __restrict__ beta)
{
  __shared__ float t[LL];
  __shared__ float rs[256];
  __shared__ float rs2[256];
  const int row = blockIdx.x;
  const int c   = row % CC;
  const float* yb = y + (long long)row * LL;
  const float* rb = r + (long long)row * LL;
  float s = 0.f, s2 = 0.f;
  for (int l = threadIdx.x; l < LL; l += blockDim.x) {
    float v = yb[l] + rb[l];
    t[l] = v; s += v; s2 += v * v;
  }
  rs[threadIdx.x] = s; rs2[threadIdx.x] = s2;
  __syncthreads();
  for (int off = 128; off > 0; off >>= 1) {
    if ((int)threadIdx.x < off) {
      rs[threadIdx.x]  += rs[threadIdx.x + off];
      rs2[threadIdx.x] += rs2[threadIdx.x + off];
    }
    __syncthreads();
  }
  float mean = rs[0] / (float)LL;
  float var  = rs2[0] / (float)LL - mean * mean;
  float inv  = rsqrtf(var + EPS) * gamma[c];
  float bta  = beta[c];
  float* db = dst + (long long)row * LL;
  for (int l = threadIdx.x; l < LL; l += blockDim.x)
    db[l] = (t[l] - mean) * inv + bta;
}

// =====================================================================
extern "C" void kernel_launch(void* const* d_in, const int* in_sizes, int n_in,
                              void* d_out, int out_size, void* d_ws, size_t ws_size,
                              hipStream_t stream)
{
  (void)in_sizes; (void)n_in; (void)out_size; (void)ws_size;
  const float* x        = (const float*)d_in[0];
  const float* enc_pw   = (const float*)d_in[1];
  const float* enc_dwa  = (const float*)d_in[2];
  const float* enc_dwb  = (const float*)d_in[3];
  const float* enc_gate = (const float*)d_in[4];
  const float* q_pw  = (const float*)d_in[5];
  const float* q_dwa = (const float*)d_in[6];
  const float* q_dwb = (const float*)d_in[7];
  const float* q_gt  = (const float*)d_in[8];
  const float* k_pw  = (const float*)d_in[9];
  const float* k_dwa = (const float*)d_in[10];
  const float* k_dwb = (const float*)d_in[11];
  const float* k_gt  = (const float*)d_in[12];
  const float* v_pw  = (const float*)d_in[13];
  const float* v_dwa = (const float*)d_in[14];
  const float* v_dwb = (const float*)d_in[15];
  const float* v_gt  = (const float*)d_in[16];
  const float* uh_w  = (const float*)d_in[17];
  const float* uh_b  = (const float*)d_in[18];
  const float* n1_g  = (const float*)d_in[19];
  const float* n1_b  = (const float*)d_in[20];
  const float* ff1_w = (const float*)d_in[21];
  const float* ff1_b = (const float*)d_in[22];
  const float* ff2_w = (const float*)d_in[23];
  const float* ff2_b = (const float*)d_in[24];
  const float* n2_g  = (const float*)d_in[25];
  const float* n2_b  = (const float*)d_in[26];

  // ---- workspace carve (floats) ----
  float* ws   = (float*)d_ws;
  float* actA = ws;                                 // [B,C,L]
  float* actB = actA + (size_t)BB * CC * LL;        // [B,C,L]
  float* ybuf = actB + (size_t)BB * CC * LL;        // [B,C,L]
  float* qbuf = ybuf + (size_t)BB * CC * LL;        // [B,CH,L]
  float* kbuf = qbuf + (size_t)BB * CH * LL;
  float* vbuf = kbuf + (size_t)BB * CH * LL;
  float* attb = vbuf + (size_t)BB * CH * LL;        // [B,CH,L]
  float* ffh  = attb + (size_t)BB * CH * LL;        // [B,FF,L]
  float* scor = ffh  + (size_t)BB * FFD * LL;       // [H,L,L] (per-batch reuse)

  const float score_scale = 0.08838834764831845f;   // 1/sqrt(C): q,k each /C^0.25

  // ---- encoder DDSC ----
  encoder_ddsc<<<BB * CC, 256, 0, stream>>>(x, enc_pw, enc_dwa, enc_dwb, enc_gate, actA);

  for (int i = 0; i < DD; ++i) {
    // --- q/k/v pointwise: [CH x C] * act[b] -> [CH x L], batched over B (weight stride 0) ---
    {
      int blocks = (BB * (CH / 64) * (LL / 64) + 7) / 8;
      wmma_gemm_f32<0, 4><<<blocks, 256, 0, stream>>>(
          q_pw + (long long)i * CH * CC, actA, qbuf, nullptr,
          CH, LL, CC, CC, LL, LL, 0, (long long)CC * LL, (long long)CH * LL,
          BB, 1.f, 0);
      wmma_gemm_f32<0, 4><<<blocks, 256, 0, stream>>>(
          k_pw + (long long)i * CH * CC, actA, kbuf, nullptr,
          CH, LL, CC, CC, LL, LL, 0, (long long)CC * LL, (long long)CH * LL,
          BB, 1.f, 0);
      wmma_gemm_f32<0, 4><<<blocks, 256, 0, stream>>>(
          v_pw + (long long)i * CH * CC, actA, vbuf, nullptr,
          CH, LL, CC, CC, LL, LL, 0, (long long)CC * LL, (long long)CH * LL,
          BB, 1.f, 0);
    }
    // --- gated depthwise convs (in place) ---
    dw_conv_rows<<<BB * CH, 256, 0, stream>>>(qbuf, q_dwa + (long long)i * CH * 3,
                                              q_dwb + (long long)i * CH * 15, q_gt + i * 2);
    dw_conv_rows<<<BB * CH, 256, 0, stream>>>(kbuf, k_dwa + (long long)i * CH * 3,
                                              k_dwb + (long long)i * CH * 15, k_gt + i * 2);
    dw_conv_rows<<<BB * CH, 256, 0, stream>>>(vbuf, v_dwa + (long long)i * CH * 3,
                                              v_dwb + (long long)i * CH * 15, v_gt + i * 2);

    // --- attention per batch (score buffer reused; stream-ordered) ---
    for (int b = 0; b < BB; ++b) {
      const float* kh = kbuf + (long long)b * CH * LL;
      const float* qh = qbuf + (long long)b * CH * LL;
      const float* vh = vbuf + (long long)b * CH * LL;
      float* ah = attb + (long long)b * CH * LL;
      // dot[kl,ql] = sum_c K[c,kl]*Q[c,ql] : A = K (stored [C,L] -> transA), batched over heads
      int blocks = (HH * (LL / 64) * (LL / 64) + 7) / 8;
      wmma_gemm_f32<1, 4><<<blocks, 256, 0, stream>>>(
          kh, qh, scor, nullptr,
          LL, LL, CC, LL, LL, LL,
          (long long)CC * LL, (long long)CC * LL, (long long)LL * LL,
          HH, score_scale, 0);
      // softmax over KEY axis
      softmax_over_rows<<<HH * (LL / 16), 256, 0, stream>>>(scor);
      // att[c,ql] = sum_kl V[c,kl]*P[kl,ql]  (M=128 -> BM=2 keeps waves plentiful)
      blocks = (HH * (CC / 32) * (LL / 64) + 7) / 8;
      wmma_gemm_f32<0, 2><<<blocks, 256, 0, stream>>>(
          vh, scor, ah, nullptr,
          CC, LL, LL, LL, LL, LL,
          (long long)CC * LL, (long long)LL * LL, (long long)CC * LL,
          HH, 1.f, 0);
    }

    // --- unify heads: [C x CH] * att[b] + bias ---
    {
      int blocks = (BB * (CC / 32) * (LL / 64) + 7) / 8;
      wmma_gemm_f32<0, 2><<<blocks, 256, 0, stream>>>(
          uh_w + (long long)i * CC * CH, attb, ybuf, uh_b + i * CC,
          CC, LL, CH, CH, LL, LL, 0, (long long)CH * LL, (long long)CC * LL,
          BB, 1.f, 0);
    }
    // --- norm1: inorm(att + xin) ---
    inorm_res<<<BB * CC, 256, 0, stream>>>(ybuf, actA, actB, n1_g + i * CC, n1_b + i * CC);

    // --- FF1 (relu) ---
    {
      int blocks = (BB * (FFD / 64) * (LL / 64) + 7) / 8;
      wmma_gemm_f32<0, 4><<<blocks, 256, 0, stream>>>(
          ff1_w + (long long)i * FFD * CC, actB, ffh, ff1_b + i * FFD,
          FFD, LL, CC, CC, LL, LL, 0, (long long)CC * LL, (long long)FFD * LL,
          BB, 1.f, 1);
    }
    // --- FF2 ---
    {
      int blocks = (BB * (CC / 32) * (LL / 64) + 7) / 8;
      wmma_gemm_f32<0, 2><<<blocks, 256, 0, stream>>>(
          ff2_w + (long long)i * CC * FFD, ffh, ybuf, ff2_b + i * CC,
          CC, LL, FFD, FFD, LL, LL, 0, (long long)FFD * LL, (long long)CC * LL,
          BB, 1.f, 0);
    }
    // --- norm2: inorm(ff + out) -> back into actA for next layer ---
    inorm_res<<<BB * CC, 256, 0, stream>>>(ybuf, actB, actA, n2_g + i * CC, n2_b + i * CC);
  }

  hipMemcpyAsync(d_out, actA, (size_t)BB * CC * LL * sizeof(float),
                 hipMemcpyDeviceToDevice, stream);
}